// Fast1_74406013436169
// MI455X (gfx1250) — compile-verified
//
#include <hip/hip_runtime.h>
#include <stdint.h>

// Problem geometry (fixed by the reference)
#define T_STEPS 256
#define H 512
#define W 512
#define HW (H * W)

// Tiling
#define TILE 32
#define HALO 34              // TILE + 2
#define LSTR 36              // padded LDS row stride (floats), multiple of 4
#define NELEM (HALO * HALO)  // 1156 halo elements per tile
#define NTHREADS 512
#define NLOAD 3              // ceil(1156 / 512) async loads per thread per tile
#define NBUF 3               // triple buffer -> prefetch depth 2

// -------- address-space helpers --------
typedef __attribute__((address_space(1))) int gint_t;   // global AS int
typedef __attribute__((address_space(3))) int lint_t;   // LDS AS int

__device__ __forceinline__ void async_load_f32(const float* gsrc, float* ldst) {
#if __has_builtin(__builtin_amdgcn_global_load_async_to_lds_b32)
    __builtin_amdgcn_global_load_async_to_lds_b32(
        (gint_t*)(uintptr_t)gsrc,
        (lint_t*)(uintptr_t)(unsigned)(uintptr_t)ldst,
        0, 0);
#else
    unsigned l = (unsigned)(uintptr_t)ldst;                 // LDS byte offset (low 32 bits)
    unsigned long long g = (unsigned long long)(uintptr_t)gsrc;
    asm volatile("global_load_async_to_lds_b32 %0, %1, off"
                 :: "v"(l), "v"(g) : "memory");
#endif
}

__device__ __forceinline__ void wait_async_le6() {
#if __has_builtin(__builtin_amdgcn_s_wait_asynccnt)
    __builtin_amdgcn_s_wait_asynccnt(6);
#else
    asm volatile("s_wait_asynccnt 0x6" ::: "memory");
#endif
}
__device__ __forceinline__ void wait_async_le3() {
#if __has_builtin(__builtin_amdgcn_s_wait_asynccnt)
    __builtin_amdgcn_s_wait_asynccnt(3);
#else
    asm volatile("s_wait_asynccnt 0x3" ::: "memory");
#endif
}
__device__ __forceinline__ void wait_async_le0() {
#if __has_builtin(__builtin_amdgcn_s_wait_asynccnt)
    __builtin_amdgcn_s_wait_asynccnt(0);
#else
    asm volatile("s_wait_asynccnt 0x0" ::: "memory");
#endif
}

// -------- kernel --------
__global__ __launch_bounds__(NTHREADS)
void lif_stencil_kernel(const float* __restrict__ x,
                        const float* __restrict__ kern,
                        float* __restrict__ out) {
    __shared__ float tile[NBUF][HALO * LSTR];

    const int tid = (int)threadIdx.x;
    const int bx  = (int)blockIdx.x;
    const int by  = (int)blockIdx.y;
    const int tx  = tid & 31;   // column in tile (== lane id -> conflict-free LDS rows)
    const int ty  = tid >> 5;   // 0..15 (uniform per wave)

    // ---- per-thread async-load slots (clamped global offset + LDS offset) ----
    int g_off[NLOAD];
    int l_off[NLOAD];
#pragma unroll
    for (int j = 0; j < NLOAD; ++j) {
        int i = tid + j * NTHREADS;
        if (i >= NELEM) i -= NELEM;           // duplicates keep EXEC all-ones
        int lr = i / HALO;
        int lc = i - lr * HALO;
        int gr = by * TILE - 1 + lr;  gr = min(max(gr, 0), H - 1);
        int gc = bx * TILE - 1 + lc;  gc = min(max(gc, 0), W - 1);
        g_off[j] = gr * W + gc;
        l_off[j] = lr * LSTR + lc;
    }

    // ---- stencil weights (uniform scalar loads; cross-shaped kernel) ----
    const float kc = kern[4];
    const float ku = kern[1];
    const float kl = kern[3];
    const float kr = kern[5];
    const float kd = kern[7];

    // ---- per-pixel constants (2 pixels per thread: rows ty and ty+16) ----
    const int   gx  = bx * TILE + tx;
    const int   gy0 = by * TILE + ty;
    const int   gy1 = gy0 + 16;
    const float wl  = (gx > 0)      ? kl : 0.0f;
    const float wr  = (gx < W - 1)  ? kr : 0.0f;
    const float wu0 = (gy0 > 0)     ? ku : 0.0f;
    const float wd0 = kd;                       // gy0 <= 495, never bottom row
    const float wu1 = ku;                       // gy1 >= 16, never top row
    const float wd1 = (gy1 < H - 1) ? kd : 0.0f;

    const int c0 = (ty + 1)  * LSTR + tx + 1;   // LDS center index, pixel 0
    const int c1 = (ty + 17) * LSTR + tx + 1;   // LDS center index, pixel 1

    float st0 = 0.0f, st1 = 0.0f;               // LIF state in registers

    // ---- prologue: prefetch t=0 and t=1 ----
#pragma unroll
    for (int j = 0; j < NLOAD; ++j)
        async_load_f32(x + g_off[j], &tile[0][l_off[j]]);
#pragma unroll
    for (int j = 0; j < NLOAD; ++j)
        async_load_f32(x + (size_t)HW + g_off[j], &tile[1][l_off[j]]);

    // ---- time loop ----
#pragma unroll 1
    for (int t = 0; t < T_STEPS; ++t) {
        if (t + 2 < T_STEPS) {
            // issue prefetch of tile t+2 into the buffer last read at iter t-1
            const float* src = x + (size_t)(t + 2) * HW;
            float* dst = tile[(t + 2) % NBUF];
#pragma unroll
            for (int j = 0; j < NLOAD; ++j)
                async_load_f32(src + g_off[j], &dst[l_off[j]]);
            wait_async_le6();   // <=6 outstanding (t+1,t+2) => tile t complete
        } else if (t + 1 < T_STEPS) {
            wait_async_le3();   // only t+1 outstanding => tile t complete
        } else {
            wait_async_le0();
        }
        __syncthreads();        // all waves' tile-t data visible

        const float* tb = tile[t % NBUF];

        float z0 = kc  * tb[c0]
                 + wu0 * tb[c0 - LSTR] + wd0 * tb[c0 + LSTR]
                 + wl  * tb[c0 - 1]    + wr  * tb[c0 + 1];
        float z1 = kc  * tb[c1]
                 + wu1 * tb[c1 - LSTR] + wd1 * tb[c1 + LSTR]
                 + wl  * tb[c1 - 1]    + wr  * tb[c1 + 1];

        st0 = 0.9f * st0 + z0;
        st1 = 0.9f * st1 + z1;

        float s0 = (st0 > 2.0f) ? 1.0f : 0.0f;
        float s1 = (st1 > 2.0f) ? 1.0f : 0.0f;

        float* o = out + (size_t)t * HW;
        __builtin_nontemporal_store(s0, o + (size_t)gy0 * W + gx);
        __builtin_nontemporal_store(s1, o + (size_t)gy1 * W + gx);

        __syncthreads();        // protect buffer (t%NBUF) before refill at iter t+1
    }
}

// -------- host launch --------
extern "C" void kernel_launch(void* const* d_in, const int* in_sizes, int n_in,
                              void* d_out, int out_size, void* d_ws, size_t ws_size,
                              hipStream_t stream) {
    (void)in_sizes; (void)n_in; (void)out_size; (void)d_ws; (void)ws_size;
    const float* x    = (const float*)d_in[0];
    const float* kern = (const float*)d_in[1];
    float*       out  = (float*)d_out;

    dim3 grid(W / TILE, H / TILE, 1);   // 16 x 16 = 256 workgroups
    dim3 block(NTHREADS, 1, 1);         // 16 wave32 waves
    hipLaunchKernelGGL(lif_stencil_kernel, grid, block, 0, stream, x, kern, out);
}